// GlobalHistogramFixedSamples_56727928045826
// MI455X (gfx1250) — compile-verified
//
#include <hip/hip_runtime.h>
#include <hip/hip_bf16.h>
#include <math.h>

#define NV 4096
#define NS 32768
#define NB 4

typedef __attribute__((ext_vector_type(2))) float v2f;
typedef __attribute__((ext_vector_type(8))) float v8f;

// ---------------------------------------------------------------------------
// Kernel 1: per-batch sigma (std ddof=1, mean over 3 comps, clipped) -> params
// ws layout (floats): [b*4+0]=scale (cn*log2e/sigma^2), [b*4+1]=coef
//                     [16+b]   = running K_H total (zeroed here)
// ---------------------------------------------------------------------------
__global__ void __launch_bounds__(256) kde_sigma_kernel(
    const float* __restrict__ T, float* __restrict__ ws, float cf, float cn) {
  const int b = blockIdx.x;
  const int t = threadIdx.x;
  const float* Tb = T + (size_t)b * NV * 3;

  float sx[3] = {0.f, 0.f, 0.f};
  float sxx[3] = {0.f, 0.f, 0.f};
  for (int v = t; v < NV; v += 256) {
    #pragma unroll
    for (int d = 0; d < 3; ++d) {
      float x = Tb[v * 3 + d];
      sx[d] += x;
      sxx[d] += x * x;
    }
  }

  __shared__ float red[6 * 256];
  #pragma unroll
  for (int k = 0; k < 3; ++k) {
    red[k * 256 + t] = sx[k];
    red[(k + 3) * 256 + t] = sxx[k];
  }
  __syncthreads();
  for (int s = 128; s > 0; s >>= 1) {
    if (t < s) {
      #pragma unroll
      for (int k = 0; k < 6; ++k) red[k * 256 + t] += red[k * 256 + t + s];
    }
    __syncthreads();
  }

  if (t == 0) {
    float sigma = 0.f;
    #pragma unroll
    for (int k = 0; k < 3; ++k) {
      float mean = red[k * 256] * (1.0f / NV);
      float var = (red[(k + 3) * 256] - (float)NV * mean * mean) * (1.0f / (NV - 1));
      sigma += sqrtf(fmaxf(var, 0.f));
    }
    sigma *= (1.0f / 3.0f);
    sigma = fmaxf(sigma, 0.1f);
    const float log2e = 1.4426950408889634f;
    float scale = cn * log2e / (sigma * sigma);
    float coef = cf / (sigma * sigma * sigma * (float)NV);
    ws[b * 4 + 0] = scale;
    ws[b * 4 + 1] = coef;
    ws[16 + b] = 0.f;
  }
}

// ---------------------------------------------------------------------------
// Kernel 2: main KDE via V_WMMA_F32_16X16X4_F32.
// grid = (NS/128, NB), block = 256 (8 wave32s). Each wave owns 16 grid points.
// LDS holds packed B operands for the whole batch: P[v] = float4(
//   -2*scale*tx, -2*scale*ty, -2*scale*tz, scale*|t|^2 )
// WMMA computes D[m][n] = scale*(|s|^2 + |t|^2 - 2 s.t)  (log2-domain exponent)
// ---------------------------------------------------------------------------
__global__ void __launch_bounds__(256) kde_main_kernel(
    const float* __restrict__ T, const float* __restrict__ S,
    const float* __restrict__ ws_params, float* __restrict__ out,
    float* __restrict__ totals) {
  __shared__ __align__(16) float sP[NV * 4];  // 64 KB

  const int b = blockIdx.y;
  const int tid = threadIdx.x;
  const float scale = ws_params[b * 4 + 0];
  const float coef = ws_params[b * 4 + 1];
  const float m2s = -2.0f * scale;

  // Stage packed B operands for the entire batch into LDS.
  const float* Tb = T + (size_t)b * NV * 3;
  #pragma unroll
  for (int i = 0; i < 16; ++i) {
    int v = tid + i * 256;
    float x = Tb[v * 3 + 0];
    float y = Tb[v * 3 + 1];
    float z = Tb[v * 3 + 2];
    float4 p;
    p.x = m2s * x;
    p.y = m2s * y;
    p.z = m2s * z;
    p.w = scale * (x * x + y * y + z * z);
    *(float4*)(sP + v * 4) = p;
  }
  __syncthreads();

  const int wave = tid >> 5;
  const int lane = tid & 31;
  const int n = lane & 15;      // N index within tile / M index for A rows
  const int khalf = lane >> 4;  // which K-pair this half-wave holds
  const int sbase = blockIdx.x * 128 + wave * 16;

  // A operand: row m = n holds (x,y,z,1); VGPR0/1 carry K = 2*khalf, 2*khalf+1
  float x = S[(sbase + n) * 3 + 0];
  float y = S[(sbase + n) * 3 + 1];
  float z = S[(sbase + n) * 3 + 2];
  v2f a;
  a.x = khalf ? z : x;
  a.y = khalf ? 1.0f : y;

  // C operand: scale*|s|^2 broadcast along N; VGPR r holds row M = r + 8*khalf
  float s2 = scale * (x * x + y * y + z * z);
  v8f cmat;
  #pragma unroll
  for (int r = 0; r < 8; ++r) cmat[r] = __shfl(s2, r + 8 * khalf, 32);

  float acc[8];
  #pragma unroll
  for (int r = 0; r < 8; ++r) acc[r] = 0.f;

  // Per-lane LDS base: element (j*16 + n), floats [2*khalf, 2*khalf+1]
  const float* lbase = sP + n * 4 + khalf * 2;

  #pragma unroll 4
  for (int j = 0; j < NV / 16; ++j) {
    v2f bv = *(const v2f*)(lbase + j * 64);
    v8f d = __builtin_amdgcn_wmma_f32_16x16x4_f32(
        false, a, false, bv, (short)0, cmat, false, false);
    #pragma unroll
    for (int r = 0; r < 8; ++r) acc[r] += __builtin_amdgcn_exp2f(d[r]);
  }

  // Reduce over N (16 lanes within each half-wave).
  #pragma unroll
  for (int m = 1; m < 16; m <<= 1) {
    #pragma unroll
    for (int r = 0; r < 8; ++r) acc[r] += __shfl_xor(acc[r], m, 32);
  }

  if (n == 0) {  // lanes 0 (rows 0..7) and 16 (rows 8..15)
    float tot = 0.f;
    #pragma unroll
    for (int r = 0; r < 8; ++r) {
      float val = coef * acc[r];
      out[(size_t)b * NS + sbase + khalf * 8 + r] = val;
      tot += val;
    }
    atomicAdd(&totals[b], tot);
  }
}

// ---------------------------------------------------------------------------
// Kernel 3: normalize by per-batch sum.
// ---------------------------------------------------------------------------
__global__ void __launch_bounds__(256) kde_norm_kernel(
    float* __restrict__ out, const float* __restrict__ totals) {
  int i = blockIdx.x * 256 + threadIdx.x;
  int b = i >> 15;  // / NS
  out[i] /= fmaxf(totals[b], 1e-5f);
}

extern "C" void kernel_launch(void* const* d_in, const int* in_sizes, int n_in,
                              void* d_out, int out_size, void* d_ws, size_t ws_size,
                              hipStream_t stream) {
  const float* T = (const float*)d_in[0];  // (B, NV, 3)
  const float* S = (const float*)d_in[1];  // (NS, 3)
  float* out = (float*)d_out;              // (B, NS)
  float* ws = (float*)d_ws;

  const double d = 3.0;
  const double c = pow(4.0 / (d + 2.0), 1.0 / (d + 4.0)) *
                   pow((double)NV, -1.0 / (d + 4.0));
  const double cf = pow(2.0 * M_PI, -1.5) / (c * c * c);
  const double cn = -1.0 / (2.0 * c * c);

  kde_sigma_kernel<<<NB, 256, 0, stream>>>(T, ws, (float)cf, (float)cn);

  dim3 grid(NS / 128, NB);
  kde_main_kernel<<<grid, 256, 0, stream>>>(T, S, ws, out, ws + 16);

  kde_norm_kernel<<<(NB * NS) / 256, 256, 0, stream>>>(out, ws + 16);
}